// SAGE_29411936043240
// MI455X (gfx1250) — compile-verified
//
#include <hip/hip_runtime.h>
#include <hip/hip_bf16.h>
#include <math.h>

// ---------------------------------------------------------------------------
// GraphSAGE 2-layer forward on gfx1250.
//   Aggregation: bandwidth-bound edge scatter with GLOBAL_ATOMIC_ADD_F32
//   (one wave32 per edge -> fully coalesced 512B/1024B feature rows).
//   Linear layers: V_WMMA_F32_16X16X4_F32, one 16x16 output tile per wave,
//   fused mean-normalize + lin_l + lin_r + bias (+ReLU).
//   Final: wave-per-row log_softmax over 64 cols.
// ---------------------------------------------------------------------------

typedef __attribute__((ext_vector_type(2))) float v2f;
typedef __attribute__((ext_vector_type(8))) float v8f;

#define N0_C 512000
#define N1_C 20480
#define N2_C 2048
#define F_IN_C 128
#define F_HID_C 256
#define F_OUT_C 64

// ---------------- zero scratch -------------------------------------------
__global__ void sage_zero_kernel(float* __restrict__ p, unsigned long long n) {
  unsigned long long i = (unsigned long long)blockIdx.x * blockDim.x + threadIdx.x;
  if (i < n) p[i] = 0.0f;
}

// ---------------- edge scatter-add (one wave per edge) -------------------
template <int F>
__global__ void sage_scatter_kernel(const float* __restrict__ feat,
                                    const int* __restrict__ src,
                                    const int* __restrict__ dst,
                                    float* __restrict__ agg,
                                    float* __restrict__ cnt,
                                    int E) {
  int wid  = (int)((blockIdx.x * blockDim.x + threadIdx.x) >> 5);
  int lane = (int)(threadIdx.x & 31);
  if (wid >= E) return;
  int s = src[wid];
  int d = dst[wid];
  const float* frow = feat + (unsigned long long)s * F;
  float*       arow = agg  + (unsigned long long)d * F;
#pragma unroll
  for (int i = 0; i < F / 32; ++i) {
    // coalesced across the wave; hardware fp32 atomic (GLOBAL_ATOMIC_ADD_F32)
    unsafeAtomicAdd(&arow[i * 32 + lane], frow[i * 32 + lane]);
  }
  if (lane == 0) unsafeAtomicAdd(&cnt[d], 1.0f);
}

// ---------------- fused SAGE linear: relu?(mean@Wl + bl + xdst@Wr) -------
// One wave computes one 16x16 output tile; K stepped by 4 via
// V_WMMA_F32_16X16X4_F32 (exact fp32).
//
// A 16x4 layout : lanes 0-15 hold M=lane, K={k,k+1}; lanes 16-31 hold
//                 M=lane-16, K={k+2,k+3}.  (CDNA5 ISA "32-bit A 16x4")
// B 4x16 layout : VGPR v, lanes 0-15 -> K=v, N=lane; lanes 16-31 -> K=v+2.
// C/D layout    : N = lane&15, M = (lane<16?0:8) + vgpr.
__global__ void sage_gemm_kernel(const float* __restrict__ agg,
                                 const float* __restrict__ cnt,
                                 const float* __restrict__ xdst,
                                 const float* __restrict__ Wl,
                                 const float* __restrict__ bl,
                                 const float* __restrict__ Wr,
                                 float* __restrict__ out,
                                 int M, int N, int K, int doRelu) {
  int lane = (int)(threadIdx.x & 31);
  int tile = (int)(blockIdx.x * (blockDim.x >> 5) + (threadIdx.x >> 5));
  int ntN  = N >> 4;
  int total = (M >> 4) * ntN;
  if (tile >= total) return;  // wave-uniform: EXEC stays all-ones for WMMA

  int mt = tile / ntN;
  int nt = tile - mt * ntN;
  int row0 = mt << 4;
  int col0 = nt << 4;

  int r   = lane & 15;               // A row / D column index within tile
  int kb  = (lane < 16) ? 0 : 2;     // K sub-offset for this lane group
  int col = col0 + r;

  float invc = 1.0f / fmaxf(cnt[row0 + r], 1.0f);

  const float* arow = agg  + (unsigned long long)(row0 + r) * K;
  const float* xrow = xdst + (unsigned long long)(row0 + r) * K;

  v8f acc = {0.f, 0.f, 0.f, 0.f, 0.f, 0.f, 0.f, 0.f};

  // mean @ Wl
  for (int kk = 0; kk < K; kk += 4) {
    v2f a, b;
    a.x = arow[kk + kb]     * invc;
    a.y = arow[kk + kb + 1] * invc;
    b.x = Wl[(unsigned long long)(kk + kb)     * N + col];
    b.y = Wl[(unsigned long long)(kk + kb + 1) * N + col];
    acc = __builtin_amdgcn_wmma_f32_16x16x4_f32(false, a, false, b,
                                                (short)0, acc, false, false);
  }
  // x_dst @ Wr
  for (int kk = 0; kk < K; kk += 4) {
    v2f a, b;
    a.x = xrow[kk + kb];
    a.y = xrow[kk + kb + 1];
    b.x = Wr[(unsigned long long)(kk + kb)     * N + col];
    b.y = Wr[(unsigned long long)(kk + kb + 1) * N + col];
    acc = __builtin_amdgcn_wmma_f32_16x16x4_f32(false, a, false, b,
                                                (short)0, acc, false, false);
  }

  float bias = bl[col];
  int rowbase = row0 + ((lane < 16) ? 0 : 8);
#pragma unroll
  for (int v = 0; v < 8; ++v) {
    float val = acc[v] + bias;
    if (doRelu) val = fmaxf(val, 0.0f);
    out[(unsigned long long)(rowbase + v) * N + col] = val;
  }
}

// ---------------- row-wise log_softmax over 64 columns -------------------
__global__ void sage_logsoftmax_kernel(const float* __restrict__ o,
                                       float* __restrict__ out, int rows) {
  int wid  = (int)((blockIdx.x * blockDim.x + threadIdx.x) >> 5);
  int lane = (int)(threadIdx.x & 31);
  if (wid >= rows) return;
  const float* rp = o + (unsigned long long)wid * F_OUT_C;
  float a = rp[lane];
  float b = rp[lane + 32];
  float m = fmaxf(a, b);
#pragma unroll
  for (int off = 16; off > 0; off >>= 1) m = fmaxf(m, __shfl_xor(m, off, 32));
  float s = __expf(a - m) + __expf(b - m);
#pragma unroll
  for (int off = 16; off > 0; off >>= 1) s += __shfl_xor(s, off, 32);
  float ls = m + __logf(s);
  float* op = out + (unsigned long long)wid * F_OUT_C;
  op[lane]      = a - ls;
  op[lane + 32] = b - ls;
}

// ---------------------------------------------------------------------------
extern "C" void kernel_launch(void* const* d_in, const int* in_sizes, int n_in,
                              void* d_out, int out_size, void* d_ws, size_t ws_size,
                              hipStream_t stream) {
  const float* x       = (const float*)d_in[0];
  const int*   ei0_src = (const int*)d_in[1];
  const int*   ei0_dst = (const int*)d_in[2];
  const int*   ei1_src = (const int*)d_in[3];
  const int*   ei1_dst = (const int*)d_in[4];
  // d_in[5]=n1, d_in[6]=n2 (scalars; compile-time constants here)
  const float* Wl0 = (const float*)d_in[7];
  const float* bl0 = (const float*)d_in[8];
  const float* Wr0 = (const float*)d_in[9];
  const float* Wl1 = (const float*)d_in[10];
  const float* bl1 = (const float*)d_in[11];
  const float* Wr1 = (const float*)d_in[12];

  const int E0 = in_sizes[1];   // 512000
  const int E1 = in_sizes[3];   // 20480

  // workspace layout (floats): [agg0][cnt0][agg1][cnt1][h][o]
  float* ws    = (float*)d_ws;
  float* agg0  = ws;                                   // N1*F_IN
  float* cnt0  = agg0 + (size_t)N1_C * F_IN_C;         // N1
  float* agg1  = cnt0 + N1_C;                          // N2*F_HID
  float* cnt1  = agg1 + (size_t)N2_C * F_HID_C;        // N2
  float* h     = cnt1 + N2_C;                          // N1*F_HID
  float* obuf  = h + (size_t)N1_C * F_HID_C;           // N2*F_OUT

  // 1. zero the accumulation region ([agg0..cnt1] is contiguous)
  unsigned long long nz =
      (unsigned long long)N1_C * F_IN_C + N1_C +
      (unsigned long long)N2_C * F_HID_C + N2_C;
  sage_zero_kernel<<<(unsigned)((nz + 255) / 256), 256, 0, stream>>>(agg0, nz);

  // 2. layer-0 edge scatter (wave per edge)
  {
    int wavesPerBlock = 8;
    int blocks = (E0 + wavesPerBlock - 1) / wavesPerBlock;
    sage_scatter_kernel<F_IN_C><<<blocks, 256, 0, stream>>>(
        x, ei0_src, ei0_dst, agg0, cnt0, E0);
  }

  // 3. layer-0 fused SAGE linear + ReLU -> h [N1, F_HID]
  {
    int tiles = (N1_C / 16) * (F_HID_C / 16);          // 20480 tiles
    int blocks = (tiles + 7) / 8;
    sage_gemm_kernel<<<blocks, 256, 0, stream>>>(
        agg0, cnt0, /*xdst=*/x, Wl0, bl0, Wr0, h,
        N1_C, F_HID_C, F_IN_C, /*doRelu=*/1);
  }

  // 4. layer-1 edge scatter
  {
    int blocks = (E1 + 7) / 8;
    sage_scatter_kernel<F_HID_C><<<blocks, 256, 0, stream>>>(
        h, ei1_src, ei1_dst, agg1, cnt1, E1);
  }

  // 5. layer-1 fused SAGE linear -> obuf [N2, F_OUT]
  {
    int tiles = (N2_C / 16) * (F_OUT_C / 16);          // 512 tiles
    int blocks = (tiles + 7) / 8;
    sage_gemm_kernel<<<blocks, 256, 0, stream>>>(
        agg1, cnt1, /*xdst=*/h, Wl1, bl1, Wr1, obuf,
        N2_C, F_OUT_C, F_HID_C, /*doRelu=*/0);
  }

  // 6. log_softmax rows -> d_out
  {
    int blocks = (N2_C + 7) / 8;
    sage_logsoftmax_kernel<<<blocks, 256, 0, stream>>>(
        obuf, (float*)d_out, N2_C);
  }
}